// GroupHadamardLayer_59562606461401
// MI455X (gfx1250) — compile-verified
//
#include <hip/hip_runtime.h>
#include <stdint.h>

// ---------------------------------------------------------------------------
// out[b] = sum_f x[b,f] * w_eff[f];  w_eff scatter-built from gc*diag*fc.
// B=16384, F=2048, f32 (128 MB streamed once -> pure bandwidth, ~5.5us @ 23.3TB/s).
// GEMV via V_WMMA_F32_16X16X4_F32 (MAC + cross-lane reduce in one op),
// async global->LDS staging (ASYNCcnt), 3-deep pipeline, 2-way K-split.
// ---------------------------------------------------------------------------

typedef __attribute__((ext_vector_type(2))) float v2f;
typedef __attribute__((ext_vector_type(8))) float v8f;

#define B_ROWS 16384
#define F_DIM  2048
#define N_PAIRS 2048            // 256 groups * 8
#define KSPLIT 2
#define KCOLS (F_DIM / KSPLIT)  // 1024 columns per wave
#define KT 64                   // columns per staged tile
#define NT (KCOLS / KT)         // 16 tiles
#define NBUF 3                  // async pipeline depth
#define PADS 68                 // padded LDS row stride (floats): 68%64=4 -> conflict-free
#define WAVES_PER_BLOCK 4
#define ROWS_PER_WAVE 16
#define ROWS_PER_BLOCK (WAVES_PER_BLOCK * ROWS_PER_WAVE)   // 64

// ---- Kernel 1: zero out[] everywhere; block 0 also builds w_eff -----------
__global__ __launch_bounds__(256)
void setup_kernel(const float* __restrict__ gc_w,
                  const float* __restrict__ diag_w,
                  const float* __restrict__ fc_w,
                  const int*   __restrict__ gidx,
                  float*       __restrict__ w_eff,
                  float*       __restrict__ out) {
    const int tid = threadIdx.x;
    out[blockIdx.x * 256 + tid] = 0.0f;            // 64 blocks x 256 = 16384
    if (blockIdx.x == 0) {
        for (int i = tid; i < F_DIM; i += 256) w_eff[i] = 0.0f;
        __syncthreads();
        for (int p = tid; p < N_PAIRS; p += 256) {
            const int n = p >> 3;
            const float wt = gc_w[p] * diag_w[n] * fc_w[n];
            atomicAdd(&w_eff[gidx[p]], wt);
        }
    }
}

// ---- CDNA5 async global->LDS copy (ASYNCcnt tracked) ----------------------
__device__ __forceinline__ void async_load_b128(uint32_t lds_byte_addr,
                                                const float* gptr) {
    asm volatile("global_load_async_to_lds_b128 %0, %1, off"
                 :
                 : "v"(lds_byte_addr),
                   "v"((unsigned long long)(uintptr_t)gptr)
                 : "memory");
}
#define WAIT_ASYNC_LE(n) asm volatile("s_wait_asynccnt " #n ::: "memory")

// ---- Kernel 2: GEMV via V_WMMA_F32_16X16X4_F32 ----------------------------
__global__ __launch_bounds__(128)
void gemv_wmma_kernel(const float* __restrict__ x,
                      const float* __restrict__ w_eff,
                      float*       __restrict__ out) {
    __shared__ __align__(16) float w_lds[KCOLS];                                    // 4 KB
    __shared__ __align__(16) float xtile[WAVES_PER_BLOCK][NBUF][ROWS_PER_WAVE * PADS]; // ~51 KB

    const int tid   = threadIdx.x;
    const int lane  = tid & 31;
    const int wave  = tid >> 5;
    const int half  = lane >> 4;        // A/B/D layout half (K pair / M half)
    const int lrow  = lane & 15;
    const int rb    = blockIdx.x >> 1;          // row-block id (0..255)
    const int ks    = blockIdx.x & 1;           // K-split id  (0..1)
    const long row0 = ((long)rb * WAVES_PER_BLOCK + wave) * ROWS_PER_WAVE;
    const int kbase = ks * KCOLS;

    // Stage this block's 1024-float slice of w_eff: 2 x b128 per thread.
    {
        const int b = tid * 4;                  // 0..508
        *(float4*)&w_lds[b]       = *(const float4*)&w_eff[kbase + b];
        *(float4*)&w_lds[b + 512] = *(const float4*)&w_eff[kbase + b + 512];
    }
    __syncthreads();

    // Async copy mapping: instr i covers rows 2i,2i+1 of the 16x64 tile.
    // lane -> row = 2i + (lane>>4), col = (lane&15)*4  (16 lanes * 16B = 256B/row).
    const int arow = lane >> 4;             // 0..1
    const int acol = (lane & 15) * 4;       // 0..60
    const float* gbase = x + (row0 + arow) * (long)F_DIM + kbase + acol;

    // LDS byte addresses for the async engine (plain u32 values, no pointers).
    uint32_t ldsb[NBUF];
#pragma unroll
    for (int bfi = 0; bfi < NBUF; ++bfi)
        ldsb[bfi] = (uint32_t)(uintptr_t)(&xtile[wave][bfi][0] + arow * PADS + acol);

    v8f acc0 = {0.f, 0.f, 0.f, 0.f, 0.f, 0.f, 0.f, 0.f};
    v8f acc1 = {0.f, 0.f, 0.f, 0.f, 0.f, 0.f, 0.f, 0.f};

    // Prologue: issue tiles 0 and 1 (16 b128 async ops = 8 KB in flight).
#pragma unroll
    for (int it = 0; it < 2; ++it)
#pragma unroll
        for (int i = 0; i < 8; ++i)
            async_load_b128(ldsb[it] + (uint32_t)(i * 2 * PADS * 4),
                            gbase + (long)(2 * i) * F_DIM + it * KT);

    int cbuf = 0;      // buffer holding tile t
    int ibuf = 2;      // buffer receiving tile t+2
    for (int t = 0; t < NT; ++t) {
        if (t + 2 < NT) {                       // uniform branch, EXEC stays full
            const uint32_t ldsn = ldsb[ibuf];
            const float* gb = gbase + (t + 2) * KT;
#pragma unroll
            for (int i = 0; i < 8; ++i)
                async_load_b128(ldsn + (uint32_t)(i * 2 * PADS * 4),
                                gb + (long)(2 * i) * F_DIM);
        }
        // Tile t complete when <= 8*(#tiles younger than t) ops remain.
        if (t <= NT - 3)      { WAIT_ASYNC_LE(16); }
        else if (t == NT - 2) { WAIT_ASYNC_LE(8);  }
        else                  { WAIT_ASYNC_LE(0);  }

        const int cb = t * KT;                  // local offset within w_lds
        // Direct indexing of the __shared__ array (variable cbuf) keeps
        // addrspace(3) inference intact -> ds_load_b64, not flat.
#pragma unroll
        for (int j = 0; j < KT / 4; j += 2) {
            const int k2a = j * 4 + 2 * half;
            const int k2b = k2a + 4;
            v2f a0 = *(const v2f*)&xtile[wave][cbuf][lrow * PADS + k2a];
            v2f b0 = *(const v2f*)&w_lds[cb + k2a];
            v2f a1 = *(const v2f*)&xtile[wave][cbuf][lrow * PADS + k2b];
            v2f b1 = *(const v2f*)&w_lds[cb + k2b];
#if defined(__has_builtin) && __has_builtin(__builtin_amdgcn_wmma_f32_16x16x4_f32)
            acc0 = __builtin_amdgcn_wmma_f32_16x16x4_f32(
                false, a0, false, b0, (short)0, acc0, false, false);
            acc1 = __builtin_amdgcn_wmma_f32_16x16x4_f32(
                false, a1, false, b1, (short)0, acc1, false, false);
#else
            {   // VALU fallback with identical D-layout semantics
                float p0 = a0.x * b0.x + a0.y * b0.y;
                float p1 = a1.x * b1.x + a1.y * b1.y;
#pragma unroll
                for (int v = 0; v < 8; ++v) {
                    const int M = v + 8 * half;
                    acc0[v] += __shfl(p0, M, 32) + __shfl(p0, M + 16, 32);
                    acc1[v] += __shfl(p1, M, 32) + __shfl(p1, M + 16, 32);
                }
            }
#endif
        }
        cbuf = (cbuf + 1 == NBUF) ? 0 : cbuf + 1;
        ibuf = (ibuf + 1 == NBUF) ? 0 : ibuf + 1;
    }

    const v8f acc = acc0 + acc1;
    // D layout: VGPR v, lanes 0-15 -> M=v, lanes 16-31 -> M=v+8; all N equal.
    // Two K-split partials per row; 2 commutative float adds -> deterministic.
    if (lrow == 0) {
#pragma unroll
        for (int v = 0; v < 8; ++v)
            atomicAdd(&out[row0 + 8 * half + v], acc[v]);
    }
}

// ---------------------------------------------------------------------------
extern "C" void kernel_launch(void* const* d_in, const int* in_sizes, int n_in,
                              void* d_out, int out_size, void* d_ws, size_t ws_size,
                              hipStream_t stream) {
    (void)in_sizes; (void)n_in; (void)out_size; (void)ws_size;
    const float* x      = (const float*)d_in[0];
    const int*   gidx   = (const int*)  d_in[1];
    const float* gc_w   = (const float*)d_in[2];
    const float* diag_w = (const float*)d_in[3];
    const float* fc_w   = (const float*)d_in[4];
    float* out   = (float*)d_out;
    float* w_eff = (float*)d_ws;   // 2048 floats of scratch

    setup_kernel<<<B_ROWS / 256, 256, 0, stream>>>(gc_w, diag_w, fc_w, gidx, w_eff, out);
    gemv_wmma_kernel<<<(B_ROWS / ROWS_PER_BLOCK) * KSPLIT, 128, 0, stream>>>(x, w_eff, out);
}